// MS_24086176596062
// MI455X (gfx1250) — compile-verified
//
#include <hip/hip_runtime.h>

// Mean-shift (B=4, D=32, N=64*64=4096), 3 iterations, fused per-iteration kernel.
// Uses V_WMMA_F32_16X16X4_F32 for both the Gram (K-tile) and the Y@K GEMM,
// exploiting symmetry of K so column sums == row sums of the tiles we compute.

typedef __attribute__((ext_vector_type(2))) float v2f;
typedef __attribute__((ext_vector_type(8))) float v8f;

#define DD   32
#define NPIX 4096
#define NBLK (NPIX / 16)
#define LDS_STRIDE 18   // padded: 18*lq mod 64 distinct for lq=0..15, keeps 8B alignment

__device__ __forceinline__ v8f wmma_f32(v2f a, v2f b, v8f c) {
  // D = A(16x4 f32) x B(4x16 f32) + C(16x16 f32)
  return __builtin_amdgcn_wmma_f32_16x16x4_f32(
      /*neg_a=*/false, a, /*neg_b=*/false, b,
      /*c_mod=*/(short)0, c, /*reuse_a=*/false, /*reuse_b=*/false);
}

__global__ __launch_bounds__(32)
void MS_meanshift_iter(const float* __restrict__ src, float* __restrict__ dst) {
  __shared__ __align__(16) float ldsE[16 * LDS_STRIDE];

  const int lane = threadIdx.x;        // wave32: one wave per block
  const int lq   = lane & 15;
  const int half = lane >> 4;
  const int b    = blockIdx.x >> 8;    // 256 m-blocks per batch
  const int mb   = blockIdx.x & 255;

  const float* Y = src + b * DD * NPIX;
  float*       O = dst + b * DD * NPIX;

  const float BW_LOG2E = 0.14426950408889634f;  // 0.1 * log2(e)

  // Gram A operand (persistent): A[i=m, k=d] = Y[d, m_i]
  // A layout 16x4: vgpr v, lane l -> (M = l%16, K = v + 2*(l/16)) within the step
  const int mcol = mb * 16 + lq;
  v2f am[8];
#pragma unroll
  for (int k = 0; k < 8; ++k) {
    am[k].x = Y[(4 * k + 0 + 2 * half) * NPIX + mcol];
    am[k].y = Y[(4 * k + 1 + 2 * half) * NPIX + mcol];
  }

  v8f acc0 = {};  // OutT tile: M = m-local, N = d (0..15)
  v8f acc1 = {};  // d (16..31)
  v8f srow = {};  // per-lane partial row sums of E (lane's n-column = lq)

  for (int nb = 0; nb < NBLK; ++nb) {
    const int ncol0 = nb * 16;
    if (nb + 1 < NBLK)  // pull next n-block toward L2/L0 (global_prefetch_b8)
      __builtin_prefetch(&Y[ncol0 + 16 + lq], 0, 0);

    // ---- Gram tile: G[i=m, j=n] = sum_d Y[d,m_i] * Y[d,n_j] ----
    // B operand: B[k=d, j=n] = Y[d, n_j]; vgpr v, lane l -> (K = v+2*(l/16), N = l%16)
    v2f gb[8];
#pragma unroll
    for (int k = 0; k < 8; ++k) {
      gb[k].x = Y[(4 * k + 0 + 2 * half) * NPIX + ncol0 + lq];
      gb[k].y = Y[(4 * k + 1 + 2 * half) * NPIX + ncol0 + lq];
    }
    v8f g = {};
#pragma unroll
    for (int k = 0; k < 8; ++k) g = wmma_f32(am[k], gb[k], g);

    // ---- E = exp(0.1*G) = exp2(G * 0.1*log2e); accumulate row sums ----
    v8f e;
#pragma unroll
    for (int r = 0; r < 8; ++r) e[r] = __builtin_amdgcn_exp2f(g[r] * BW_LOG2E);
    srow += e;

    // ---- stage E in LDS for C-layout -> A-layout conversion ----
    __syncthreads();  // single-wave WG: S_NOP + compiler DS fence
#pragma unroll
    for (int r = 0; r < 8; ++r)
      ldsE[(r + 8 * half) * LDS_STRIDE + lq] = e[r];  // ldsE[m_local][n_local]
    __syncthreads();

    // A2 operand: A2[i=m, k=n] = E[i][k]; lane l needs E[lq][4k + v + 2*half]
    v2f a2[4];
#pragma unroll
    for (int k = 0; k < 4; ++k)
      a2[k] = *(const v2f*)&ldsE[lq * LDS_STRIDE + 4 * k + 2 * half];

    // ---- OutT[i=m, j=d] += sum_n E[i,n] * Y[d, n]  (B2[k=n, j=d] = Y[d, n_k]) ----
#pragma unroll
    for (int k = 0; k < 4; ++k) {
      v2f b20 = *(const v2f*)&Y[(0 * 16 + lq) * NPIX + ncol0 + 4 * k + 2 * half];
      v2f b21 = *(const v2f*)&Y[(1 * 16 + lq) * NPIX + ncol0 + 4 * k + 2 * half];
      acc0 = wmma_f32(a2[k], b20, acc0);
      acc1 = wmma_f32(a2[k], b21, acc1);
    }
  }

  // ---- reduce srow across the 16 lanes of each half (butterfly) ----
#pragma unroll
  for (int mask = 1; mask <= 8; mask <<= 1) {
#pragma unroll
    for (int r = 0; r < 8; ++r)
      srow[r] += __shfl_xor(srow[r], mask, 32);
  }

  // ---- out[d, m] = 0.5 * OutT[m, d] / s[m] + 0.5 * Y[d, m] ----
  // C layout: vgpr r, lane l -> (M = r + 8*half, N = l%16)
#pragma unroll
  for (int r = 0; r < 8; ++r) {
    const int m   = mb * 16 + r + 8 * half;
    const float inv = 0.5f / srow[r];
    {
      const int d = lq;          // dblk 0
      O[d * NPIX + m] = acc0[r] * inv + 0.5f * Y[d * NPIX + m];
    }
    {
      const int d = 16 + lq;     // dblk 1
      O[d * NPIX + m] = acc1[r] * inv + 0.5f * Y[d * NPIX + m];
    }
  }
}

extern "C" void kernel_launch(void* const* d_in, const int* in_sizes, int n_in,
                              void* d_out, int out_size, void* d_ws, size_t ws_size,
                              hipStream_t stream) {
  (void)in_sizes; (void)n_in; (void)out_size; (void)ws_size;
  const float* x  = (const float*)d_in[0];      // (4, 32, 64, 64) f32
  float*       out = (float*)d_out;
  float* ws0 = (float*)d_ws;                    // needs 2 * 4*32*4096 floats = 4 MB
  float* ws1 = ws0 + 4 * DD * NPIX;

  dim3 grid(4 * NBLK);   // batch * m-blocks = 1024 waves
  dim3 block(32);        // one wave32 per workgroup

  // 3 mean-shift iterations, ping-pong through workspace
  MS_meanshift_iter<<<grid, block, 0, stream>>>(x,   ws0);
  MS_meanshift_iter<<<grid, block, 0, stream>>>(ws0, ws1);
  MS_meanshift_iter<<<grid, block, 0, stream>>>(ws1, out);
}